// StrokeParametersToImage_71622874628390
// MI455X (gfx1250) — compile-verified
//
#include <hip/hip_runtime.h>

#define BB 64
#define NN 32
#define HH 200
#define WW 200
#define NSEG (NN - 1)
#define SEG_STRIDE 12           // floats per segment record (3 x float4)
#define PIX (HH * WW)
#define PPT 4                   // pixels per thread
#define TPB 256

typedef __attribute__((ext_vector_type(2))) float v2f;
typedef __attribute__((ext_vector_type(8))) float v8f;

// ---------------------------------------------------------------------------
// Kernel A: per-batch stroke prep. 64 blocks x 1 wave (32 lanes).
// Rotates/scales the 32 control points, applies the 2x2 affine via
// V_WMMA_F32_16X16X4_F32 (exact fp32), then emits 31 segment records:
//   {vy, vx, wvy, wvx}, {A, By, Bx, th0f}, {dthf, 0, 0, 0}
// where dot = A + By*gy + Bx*gx  (1/(d2+1e-5) folded in) and
// th = th0f + frac*dthf          (thick_mult/bias folded in).
// All lane-dependent selection is done with selects (no exec-mask branches);
// D-matrix extraction stores unconditionally via a selected LDS address.
// ---------------------------------------------------------------------------
__global__ __launch_bounds__(32) void stroke_prep(
    const float* __restrict__ param,
    const float* __restrict__ x_start, const float* __restrict__ y_start,
    const float* __restrict__ theta,
    const float* __restrict__ Wm, const float* __restrict__ bias,
    const float* __restrict__ tmul, const float* __restrict__ tbias,
    const float* __restrict__ xm, const float* __restrict__ ym,
    const float* __restrict__ xb, const float* __restrict__ yb,
    float* __restrict__ seg)
{
  __shared__ float yyL[NN], xxL[NN], thL[NN];
  __shared__ float sL[NN * 2 + 16];   // [point*2+n]; +16 = dump region
  const int lane = threadIdx.x;       // 0..31, one full wave32
  const int b = blockIdx.x;

  const float p0 = param[(b * NN + lane) * 3 + 0];
  const float p1 = param[(b * NN + lane) * 3 + 1];
  const float p2 = param[(b * NN + lane) * 3 + 2];
  const float th = theta[0];
  const float c = cosf(th), s = sinf(th);
  const float x0 = p0, y0 = -p1;
  const float xr = x0 * c + y0 * s;
  const float yr = -x0 * s + y0 * c;
  yyL[lane] = yr * ym[0] + yb[0] + y_start[0];
  xxL[lane] = xr * xm[0] + xb[0] + x_start[0];
  thL[lane] = (p2 * 2.0f + 0.5f) * 0.015625f;   // /64 exact
  __syncthreads();

  const int n  = lane & 15;          // N column / M row within half
  const bool hi = lane >= 16;        // hi half supplies K=2,3 (zero pad)

  // Unconditional scalar-uniform loads, selection via cndmask only.
  const float w00 = Wm[0], w01 = Wm[1], w10 = Wm[2], w11 = Wm[3];
  const float b0 = bias[0], b1 = bias[1];

  // B matrix (4x16) = W^T padded: B[k][n] = W[n][k], rows 2,3 zero.
  v2f bmat;
  bmat.x = hi ? 0.f : ((n == 0) ? w00 : (n == 1) ? w10 : 0.f);
  bmat.y = hi ? 0.f : ((n == 0) ? w01 : (n == 1) ? w11 : 0.f);
  // C matrix: bias broadcast down each of the first two columns.
  const float cval = (n == 0) ? b0 : (n == 1) ? b1 : 0.f;
  v8f cm;
#pragma unroll
  for (int r = 0; r < 8; ++r) cm[r] = cval;

#pragma unroll
  for (int g = 0; g < 2; ++g) {      // points 0..15, then 16..31
    // Unconditional LDS loads (index always in range), then select.
    const float yv = yyL[g * 16 + n];
    const float xv = xxL[g * 16 + n];
    v2f a;                           // A row m: [yy, xx, 0, 0]
    a.x = hi ? 0.f : yv;             // K=0 (lo) / K=2 (hi, zero)
    a.y = hi ? 0.f : xv;             // K=1 (lo) / K=3 (hi, zero)
    v8f d = __builtin_amdgcn_wmma_f32_16x16x4_f32(
        false, a, false, bmat, (short)0, cm, false, false);
#pragma unroll
    for (int r = 0; r < 8; ++r) {
      const int m = g * 16 + r + (hi ? 8 : 0);      // point index
      // lanes n<2 hold columns 0/1 (= sy/sx); others write to dump region
      const int addr = (n < 2) ? (m * 2 + n) : (NN * 2 + n);
      sL[addr] = d[r];                              // unconditional ds_store
    }
  }
  __syncthreads();

  if (lane < NSEG) {
    const float vy = sL[lane * 2 + 0],       vx = sL[lane * 2 + 1];
    const float wy = sL[(lane + 1) * 2 + 0], wx = sL[(lane + 1) * 2 + 1];
    const float wvy = wy - vy, wvx = wx - vx;
    const float d2  = wvy * wvy + wvx * wvx;
    const float inv = 1.0f / (d2 + 1e-5f);
    const float A  = -(vy * wvy + vx * wvx) * inv;
    const float By = wvy * inv;
    const float Bx = wvx * inv;
    const float tm = tmul[0], tb = tbias[0];
    const float th0f = thL[lane] * tm + tb;
    const float dthf = (thL[lane + 1] - thL[lane]) * tm;
    float4* o = (float4*)(seg + (b * NSEG + lane) * SEG_STRIDE);
    o[0] = make_float4(vy, vx, wvy, wvx);
    o[1] = make_float4(A, By, Bx, th0f);
    o[2] = make_float4(dthf, 0.f, 0.f, 0.f);
  }
}

// ---------------------------------------------------------------------------
// Kernel B: rasterize. grid (40, 64) x 256 threads, 4 pixels/thread.
// Segment table staged to LDS via async global->LDS DMA; inner loop is pure
// FMA/min/max + one v_rcp_f32 per segment. The pow/sigmoid tail is applied
// once per pixel after the max (legal: the transform is monotone in dark).
// ---------------------------------------------------------------------------
__global__ __launch_bounds__(TPB) void stroke_raster(
    const float* __restrict__ seg,
    const float* __restrict__ dexp_p, const float* __restrict__ dnlw_p,
    float* __restrict__ out)
{
  __shared__ __align__(16) float segLDS[NSEG * SEG_STRIDE];
  const int b = blockIdx.y;
  const int t = threadIdx.x;

  if (t < NSEG * 3) {                               // 93 x float4 = 1488 B
    const float* gsrc = seg + b * NSEG * SEG_STRIDE + t * 4;
#if defined(__gfx1250__)
    unsigned int loff = (unsigned int)(unsigned long long)(&segLDS[t * 4]);
    unsigned long long ga = (unsigned long long)gsrc;
    asm volatile("global_load_async_to_lds_b128 %0, %1, off"
                 :: "v"(loff), "v"(ga) : "memory");
#else
    *(float4*)(&segLDS[t * 4]) = *(const float4*)gsrc;
#endif
  }
#if defined(__gfx1250__)
  asm volatile("s_wait_asynccnt 0x0" ::: "memory");
#endif
  __syncthreads();

  const int base = (blockIdx.x * TPB + t) * PPT;    // multiple of 4; PIX%4==0
  const bool live = (base < PIX);
  const int sidx = live ? base : 0;

  float gy[PPT], gx[PPT], qmin[PPT];
#pragma unroll
  for (int p = 0; p < PPT; ++p) {
    const int idx = sidx + p;
    const int h = idx / WW;
    const int w = idx - h * WW;
    gy[p] = (float)h / (float)HH;
    gx[p] = (float)w / (float)WW;
    qmin[p] = 3.0e38f;                              // min of dist^2/th^2
  }

  const float4* sv = (const float4*)segLDS;
#pragma unroll 2
  for (int j = 0; j < NSEG; ++j) {
    const float4 c0 = sv[j * 3 + 0];                // vy vx wvy wvx
    const float4 c1 = sv[j * 3 + 1];                // A  By Bx  th0f
    const float dthf = segLDS[j * SEG_STRIDE + 8];
#pragma unroll
    for (int p = 0; p < PPT; ++p) {
      float dot  = fmaf(c1.y, gy[p], fmaf(c1.z, gx[p], c1.x));
      float frac = fminf(fmaxf(dot, 0.f), 1.f);
      float ey   = (gy[p] - c0.x) - frac * c0.z;
      float ex   = (gx[p] - c0.y) - frac * c0.w;
      float d2s  = fmaf(ey, ey, ex * ex);
      float thv  = fmaxf(fmaf(frac, dthf, c1.w), 1e-8f);
      float rth  = __builtin_amdgcn_rcpf(thv);
      float q    = d2s * rth * rth;                 // (dist/th)^2
      qmin[p]    = fminf(qmin[p], q);
    }
  }

  const float e   = dexp_p[0];
  const float wnl = dnlw_p[0];
  float4 res;
  float* resp = (float*)&res;
#pragma unroll
  for (int p = 0; p < PPT; ++p) {
    const float dt   = sqrtf(qmin[p]);
    const float dark = fminf(fmaxf(1.f - dt, 0.f), 1.f);
    const float dp   = __powf(dark + 1e-4f, e);
    const float z    = fmaf(dp, 16.f, -8.f);
    const float nl   = 1.f / (1.f + __expf(-z));    // sigmoid(16*dp - 8)
    resp[p] = fmaf(wnl, nl - dp, dp);               // wnl*nl + (1-wnl)*dp
  }
  if (live) {
    *(float4*)(out + b * PIX + base) = res;         // global_store_b128
  }
}

extern "C" void kernel_launch(void* const* d_in, const int* in_sizes, int n_in,
                              void* d_out, int out_size, void* d_ws, size_t ws_size,
                              hipStream_t stream) {
  const float* param = (const float*)d_in[0];
  const float* xs = (const float*)d_in[1];
  const float* ys = (const float*)d_in[2];
  const float* th = (const float*)d_in[3];
  const float* Wm = (const float*)d_in[4];
  const float* bi = (const float*)d_in[5];
  const float* tm = (const float*)d_in[6];
  const float* tb = (const float*)d_in[7];
  const float* de = (const float*)d_in[8];
  const float* dw = (const float*)d_in[9];
  const float* xm = (const float*)d_in[10];
  const float* ym = (const float*)d_in[11];
  const float* xb = (const float*)d_in[12];
  const float* yb = (const float*)d_in[13];
  // d_in[14] = use_conv, unused by the reference math.

  float* segbuf = (float*)d_ws;                     // 64*31*12*4 = 95 KB
  float* out = (float*)d_out;

  stroke_prep<<<dim3(BB), dim3(32), 0, stream>>>(
      param, xs, ys, th, Wm, bi, tm, tb, xm, ym, xb, yb, segbuf);

  const int tiles = (PIX + TPB * PPT - 1) / (TPB * PPT);   // 40
  stroke_raster<<<dim3(tiles, BB), dim3(TPB), 0, stream>>>(segbuf, de, dw, out);
}